// LM_28157805593121
// MI455X (gfx1250) — compile-verified
//
#include <hip/hip_runtime.h>
#include <hip/hip_bf16.h>
#include <math.h>

// Problem constants from the reference
#define V_  32000
#define E_  512
#define H_  512
#define T_  128
#define B_  4
#define D_  4
#define L_  2
#define C0_ 2000
#define C1_ 10000
#define NT_ (T_ - D_ + 1)   // 125

typedef float v2f __attribute__((ext_vector_type(2)));
typedef float v8f __attribute__((ext_vector_type(8)));

// ---------------------------------------------------------------------------
// fp32 WMMA GEMM with compile-time transpose flags and 2x2 register blocking.
//   C[m][n] = sum_k opA(m,k)*opB(k,n) + bias[n]
//   opA(m,k) = TRANSA ? A[k*lda+m] : A[m*lda+k]
//   opB(k,n) = TRANSB ? B[n*ldb+k] : B[k*ldb+n]
// Each wave computes a 32x32 region of C as 2x2 WMMA 16x16 tiles
// (V_WMMA_F32_16X16X4_F32, K step 4): 4 WMMAs per 4 b64 fragment loads.
// Block = 128 threads = 4 waves spanning 128 columns; block covers 32 rows.
// Loads are clamped (garbage lands only in un-stored rows/cols); stores are
// guarded by (row < M) && (col < Nc).  K must be a multiple of 4 (true here).
// ---------------------------------------------------------------------------
__device__ __forceinline__ void store_tile(float* __restrict__ Cb, const v8f& acc,
                                           int tm, int tn, int lo, int half,
                                           int ldc, int M, int Nc,
                                           const float* __restrict__ bias)
{
    const int col = tn + lo;
    if (col < Nc) {
        const float bv = bias ? bias[col] : 0.f;
        #pragma unroll
        for (int i = 0; i < 8; ++i) {
            const int row = tm + i + half * 8;   // VGPR i: M=i (lanes0-15) / M=8+i
            if (row < M)
                Cb[(long)row * ldc + col] = acc[i] + bv;
        }
    }
}

template <int TRANSA, int TRANSB>
__global__ void gemm_wmma_f32(const float* __restrict__ A, int lda, long sAb,
                              const float* __restrict__ Bm, int ldb, long sBb,
                              const float* __restrict__ bias,
                              float* __restrict__ C, int ldc, long sCb,
                              int M, int N, int K, int Nc)
{
    const int bz    = blockIdx.z;
    const float* Ab = A  + (long)bz * sAb;
    const float* Bb = Bm + (long)bz * sBb;
    float*       Cb = C  + (long)bz * sCb;

    const int wave = threadIdx.x >> 5;
    const int lane = threadIdx.x & 31;
    const int half = lane >> 4;      // 0: lanes 0-15, 1: lanes 16-31
    const int lo   = lane & 15;

    const int tm = blockIdx.y * 32;
    const int tn = blockIdx.x * 128 + wave * 32;
    if (tm >= M) return;             // wave-uniform; EXEC stays all-ones

    // clamped row/col indices for fragment loads
    const int am0 = min(tm + lo,      M - 1);
    const int am1 = min(tm + 16 + lo, M - 1);
    const int bn0 = min(tn + lo,      N - 1);
    const int bn1 = min(tn + 16 + lo, N - 1);

    v8f acc00 = {0.f,0.f,0.f,0.f,0.f,0.f,0.f,0.f};
    v8f acc01 = acc00, acc10 = acc00, acc11 = acc00;

    #pragma unroll 2
    for (int k0 = 0; k0 < K; k0 += 4) {
        const int kk = k0 + half * 2;   // lanes 0-15: K=k0,k0+1; 16-31: K=k0+2,k0+3
        v2f a0, a1, b0, b1;
        if (TRANSA) {
            a0.x = Ab[(long)kk * lda + am0];  a0.y = Ab[(long)(kk + 1) * lda + am0];
            a1.x = Ab[(long)kk * lda + am1];  a1.y = Ab[(long)(kk + 1) * lda + am1];
        } else {
            a0 = *reinterpret_cast<const v2f*>(Ab + (long)am0 * lda + kk);
            a1 = *reinterpret_cast<const v2f*>(Ab + (long)am1 * lda + kk);
        }
        if (TRANSB) {
            b0 = *reinterpret_cast<const v2f*>(Bb + (long)bn0 * ldb + kk);
            b1 = *reinterpret_cast<const v2f*>(Bb + (long)bn1 * ldb + kk);
        } else {
            b0.x = Bb[(long)kk * ldb + bn0];  b0.y = Bb[(long)(kk + 1) * ldb + bn0];
            b1.x = Bb[(long)kk * ldb + bn1];  b1.y = Bb[(long)(kk + 1) * ldb + bn1];
        }
        acc00 = __builtin_amdgcn_wmma_f32_16x16x4_f32(false, a0, false, b0, (short)0, acc00, false, false);
        acc01 = __builtin_amdgcn_wmma_f32_16x16x4_f32(false, a0, false, b1, (short)0, acc01, false, false);
        acc10 = __builtin_amdgcn_wmma_f32_16x16x4_f32(false, a1, false, b0, (short)0, acc10, false, false);
        acc11 = __builtin_amdgcn_wmma_f32_16x16x4_f32(false, a1, false, b1, (short)0, acc11, false, false);
    }

    store_tile(Cb, acc00, tm,      tn,      lo, half, ldc, M, Nc, bias);
    store_tile(Cb, acc01, tm,      tn + 16, lo, half, ldc, M, Nc, bias);
    store_tile(Cb, acc10, tm + 16, tn,      lo, half, ldc, M, Nc, bias);
    store_tile(Cb, acc11, tm + 16, tn + 16, lo, half, ldc, M, Nc, bias);
}

// ---------------------------------------------------------------------------
// Embedding gather: embedded[r][e] = emb[x[r]][e]; also seeds input_f.
// ---------------------------------------------------------------------------
__global__ void embed_gather(const float* __restrict__ emb, const int* __restrict__ x,
                             float* __restrict__ embedded, float* __restrict__ input_f,
                             int rows)
{
    long long idx = (long long)blockIdx.x * blockDim.x + threadIdx.x;
    long long total = (long long)rows * E_;
    if (idx >= total) return;
    int r = (int)(idx / E_), e = (int)(idx % E_);
    float v = emb[(long long)x[r] * E_ + e];
    embedded[idx] = v;
    input_f[idx]  = v;
}

// ---------------------------------------------------------------------------
// GRU gate combine: hnext = (1-z)*n + z*h; optional masked write into hs.
// ---------------------------------------------------------------------------
__global__ void gru_combine(const float* __restrict__ gi, const float* __restrict__ gh,
                            const float* __restrict__ hprev, float* __restrict__ hnext,
                            float* __restrict__ hs, const int* __restrict__ lengths,
                            int M, int d)
{
    long long idx = (long long)blockIdx.x * blockDim.x + threadIdx.x;
    if (idx >= (long long)M * H_) return;
    int m = (int)(idx / H_), j = (int)(idx % H_);
    long long base = (long long)m * (3 * H_) + j;
    float ir = gi[base], iz = gi[base + H_], in_ = gi[base + 2 * H_];
    float hr = gh[base], hz = gh[base + H_], hn_ = gh[base + 2 * H_];
    float h  = hprev[idx];
    float r  = 1.f / (1.f + expf(-(ir + hr)));
    float z  = 1.f / (1.f + expf(-(iz + hz)));
    float n  = tanhf(in_ + r * hn_);
    float hn = (1.f - z) * n + z * h;
    hnext[idx] = hn;
    if (hs) {
        int b = m / NT_, t = m % NT_;
        bool valid = (t + d) < lengths[b];
        hs[((long long)m * D_ + d) * H_ + j] = valid ? hn : 0.f;
    }
}

// ---------------------------------------------------------------------------
// Window gather: wins layout [D][B*NT][E]; tok = (i==0&&d==0)? len[b]-1 : i+d-1
// ---------------------------------------------------------------------------
__global__ void wins_gather(const float* __restrict__ embedded,
                            const int* __restrict__ lengths,
                            float* __restrict__ wins)
{
    long long idx = (long long)blockIdx.x * blockDim.x + threadIdx.x;
    long long total = (long long)D_ * B_ * NT_ * E_;
    if (idx >= total) return;
    int e = (int)(idx % E_);
    long long r = idx / E_;
    int m = (int)(r % (B_ * NT_));
    int d = (int)(r / (B_ * NT_));
    int b = m / NT_, i = m % NT_;
    int tok = (i == 0 && d == 0) ? (lengths[b] - 1) : (i + d - 1);
    wins[idx] = embedded[((long long)(b * T_ + tok)) * E_ + e];
}

// decoder h0 = input_f[:, :NT, :]
__global__ void hdec_copy(const float* __restrict__ input_f, float* __restrict__ hdec)
{
    long long idx = (long long)blockIdx.x * blockDim.x + threadIdx.x;
    if (idx >= (long long)B_ * NT_ * H_) return;
    int m = (int)(idx / H_), j = (int)(idx % H_);
    int b = m / NT_, t = m % NT_;
    hdec[idx] = input_f[((long long)(b * T_ + t)) * H_ + j];
}

// ---------------------------------------------------------------------------
// Head log-softmax over 2002 logits; writes cols [0,2000) of d_out (zeroed for
// invalid rows) and saves the two cluster gate log-probs.
// ---------------------------------------------------------------------------
__global__ void head_softmax(const float* __restrict__ Z, int ldz,
                             float* __restrict__ out, float* __restrict__ g0g1,
                             const int* __restrict__ lengths)
{
    const int r = blockIdx.x;                  // 0..1999, r = m*D + d
    const float* z = Z + (long long)r * ldz;
    __shared__ float red[256];
    const int tid = threadIdx.x;

    float mx = -INFINITY;
    for (int c = tid; c < C0_ + 2; c += blockDim.x) mx = fmaxf(mx, z[c]);
    red[tid] = mx; __syncthreads();
    for (int s = 128; s > 0; s >>= 1) {
        if (tid < s) red[tid] = fmaxf(red[tid], red[tid + s]);
        __syncthreads();
    }
    mx = red[0]; __syncthreads();

    float sum = 0.f;
    for (int c = tid; c < C0_ + 2; c += blockDim.x) sum += expf(z[c] - mx);
    red[tid] = sum; __syncthreads();
    for (int s = 128; s > 0; s >>= 1) {
        if (tid < s) red[tid] += red[tid + s];
        __syncthreads();
    }
    const float lse = mx + logf(red[0]);

    const int m = r / D_;
    const int b = m / NT_, t = m % NT_;
    const bool valid = t < lengths[b];

    for (int c = tid; c < C0_; c += blockDim.x)
        out[(long long)r * V_ + c] = valid ? (z[c] - lse) : 0.f;
    if (tid == 0) {
        g0g1[2 * r]     = z[C0_]     - lse;
        g0g1[2 * r + 1] = z[C0_ + 1] - lse;
    }
}

// In-place log-softmax over an output column segment, plus head gate term.
__global__ void seg_softmax(float* __restrict__ out, int coloff, int C,
                            const float* __restrict__ g0g1, int gsel,
                            const int* __restrict__ lengths)
{
    const int r = blockIdx.x;
    float* z = out + (long long)r * V_ + coloff;
    __shared__ float red[256];
    const int tid = threadIdx.x;

    float mx = -INFINITY;
    for (int c = tid; c < C; c += blockDim.x) mx = fmaxf(mx, z[c]);
    red[tid] = mx; __syncthreads();
    for (int s = 128; s > 0; s >>= 1) {
        if (tid < s) red[tid] = fmaxf(red[tid], red[tid + s]);
        __syncthreads();
    }
    mx = red[0]; __syncthreads();

    float sum = 0.f;
    for (int c = tid; c < C; c += blockDim.x) sum += expf(z[c] - mx);
    red[tid] = sum; __syncthreads();
    for (int s = 128; s > 0; s >>= 1) {
        if (tid < s) red[tid] += red[tid + s];
        __syncthreads();
    }
    const float lse = mx + logf(red[0]);
    const float g = g0g1[2 * r + gsel];

    const int m = r / D_;
    const int b = m / NT_, t = m % NT_;
    const bool valid = t < lengths[b];

    for (int c = tid; c < C; c += blockDim.x)
        z[c] = valid ? (g + z[c] - lse) : 0.f;
}

// ---------------------------------------------------------------------------
static inline int cdiv(int a, int b) { return (a + b - 1) / b; }

extern "C" void kernel_launch(void* const* d_in, const int* in_sizes, int n_in,
                              void* d_out, int out_size, void* d_ws, size_t ws_size,
                              hipStream_t stream)
{
    (void)in_sizes; (void)n_in; (void)out_size; (void)ws_size;
    const int*   x        = (const int*)  d_in[0];
    const int*   lengths  = (const int*)  d_in[1];
    const float* G        = (const float*)d_in[2];
    const float* emb      = (const float*)d_in[3];
    const float* enc_w_ih = (const float*)d_in[4];
    const float* enc_w_hh = (const float*)d_in[5];
    const float* enc_b_ih = (const float*)d_in[6];
    const float* enc_b_hh = (const float*)d_in[7];
    const float* dec_w_ih = (const float*)d_in[8];
    const float* dec_w_hh = (const float*)d_in[9];
    const float* dec_b_ih = (const float*)d_in[10];
    const float* dec_b_hh = (const float*)d_in[11];
    const float* head_w   = (const float*)d_in[12];
    const float* t0_proj  = (const float*)d_in[13];
    const float* t0_out   = (const float*)d_in[14];
    const float* t1_proj  = (const float*)d_in[15];
    const float* t1_out   = (const float*)d_in[16];
    float* out = (float*)d_out;

    // workspace layout (floats)
    float* w = (float*)d_ws;
    const long SZ_BT_E  = (long)B_ * T_ * E_;       // 262144
    const long SZ_GATE  = (long)B_ * T_ * 3 * H_;   // 786432
    const long SZ_WINS  = (long)D_ * B_ * NT_ * E_; // 1024000
    const long SZ_HDEC  = (long)B_ * NT_ * H_;      // 256000
    const long SZ_HS    = (long)D_ * B_ * NT_ * H_; // 1024000
    const long SZ_GATED = (long)B_ * NT_ * 3 * H_;  // 768000
    const int  LDHEAD   = 2016;                     // padded 2002

    float* embedded = w;                 w += SZ_BT_E;
    float* input_f  = w;                 w += SZ_BT_E;
    float* wgt      = w;                 w += SZ_BT_E;
    float* gi       = w;                 w += SZ_GATE;
    float* gh       = w;                 w += SZ_GATE;
    float* wins     = w;                 w += SZ_WINS;
    float* hdec     = w;                 w += SZ_HDEC;
    float* hs       = w;                 w += SZ_HS;
    float* gid      = w;                 w += SZ_GATED;
    float* ghd      = w;                 w += SZ_GATED;
    float* Zhead    = w;                 w += (long)2000 * LDHEAD;
    float* mid0     = w;                 w += (long)2000 * 128;
    float* mid1     = w;                 w += (long)2000 * 32;
    float* g0g1     = w;                 w += (long)2000 * 2;

    const dim3 gblk(128);  // 4 waves, 32 rows x 128 cols of C per block

    // 1) embedding gather
    {
        long long n = (long long)B_ * T_ * E_;
        embed_gather<<<dim3((unsigned)((n + 255) / 256)), dim3(256), 0, stream>>>(
            emb, x, embedded, input_f, B_ * T_);
    }

    // 2) encoder layers
    for (int l = 0; l < L_; ++l) {
        // wgt[b] = G[b,l]^T @ input_f[b]   (M=T, N=E, K=T), batched over B
        gemm_wmma_f32<1, 0><<<dim3(cdiv(E_, 128), cdiv(T_, 32), B_), gblk, 0, stream>>>(
            G + (long)l * T_ * T_, T_, (long)L_ * T_ * T_,
            input_f, E_, (long)T_ * E_,
            nullptr, wgt, E_, (long)T_ * E_,
            T_, E_, T_, E_);
        // gi = wgt @ w_ih^T + b_ih   (M=B*T, N=3H, K=H)
        gemm_wmma_f32<0, 1><<<dim3(cdiv(3 * H_, 128), cdiv(B_ * T_, 32), 1), gblk, 0, stream>>>(
            wgt, H_, 0,
            enc_w_ih + (long)l * 3 * H_ * H_, H_, 0,
            enc_b_ih + (long)l * 3 * H_,
            gi, 3 * H_, 0, B_ * T_, 3 * H_, H_, 3 * H_);
        // gh = input_f @ w_hh^T + b_hh
        gemm_wmma_f32<0, 1><<<dim3(cdiv(3 * H_, 128), cdiv(B_ * T_, 32), 1), gblk, 0, stream>>>(
            input_f, H_, 0,
            enc_w_hh + (long)l * 3 * H_ * H_, H_, 0,
            enc_b_hh + (long)l * 3 * H_,
            gh, 3 * H_, 0, B_ * T_, 3 * H_, H_, 3 * H_);
        // gate combine (in place on input_f)
        long long n = (long long)B_ * T_ * H_;
        gru_combine<<<dim3((unsigned)((n + 255) / 256)), dim3(256), 0, stream>>>(
            gi, gh, input_f, input_f, nullptr, lengths, B_ * T_, 0);
    }

    // 3) decoder windows + h0
    {
        long long n = (long long)D_ * B_ * NT_ * E_;
        wins_gather<<<dim3((unsigned)((n + 255) / 256)), dim3(256), 0, stream>>>(
            embedded, lengths, wins);
        long long m = (long long)B_ * NT_ * H_;
        hdec_copy<<<dim3((unsigned)((m + 255) / 256)), dim3(256), 0, stream>>>(
            input_f, hdec);
    }

    // 4) decoder GRU scan, D steps (M = B*NT = 500)
    const int MD = B_ * NT_;
    for (int d = 0; d < D_; ++d) {
        gemm_wmma_f32<0, 1><<<dim3(cdiv(3 * H_, 128), cdiv(MD, 32), 1), gblk, 0, stream>>>(
            wins + (long)d * MD * E_, E_, 0,
            dec_w_ih, E_, 0, dec_b_ih,
            gid, 3 * H_, 0, MD, 3 * H_, E_, 3 * H_);
        gemm_wmma_f32<0, 1><<<dim3(cdiv(3 * H_, 128), cdiv(MD, 32), 1), gblk, 0, stream>>>(
            hdec, H_, 0,
            dec_w_hh, H_, 0, dec_b_hh,
            ghd, 3 * H_, 0, MD, 3 * H_, H_, 3 * H_);
        long long n = (long long)MD * H_;
        gru_combine<<<dim3((unsigned)((n + 255) / 256)), dim3(256), 0, stream>>>(
            gid, ghd, hdec, hdec, hs, lengths, MD, d);
    }

    // 5) adaptive log-softmax tail over hs [2000, H]
    const int NR = D_ * B_ * NT_;  // 2000 rows
    // head logits [2000, 2002] (ldc padded to 2016, stores guarded at 2002)
    gemm_wmma_f32<0, 1><<<dim3(cdiv(2048, 128), cdiv(NR, 32), 1), gblk, 0, stream>>>(
        hs, H_, 0, head_w, H_, 0, nullptr,
        Zhead, LDHEAD, 0, NR, C0_ + 2, H_, C0_ + 2);
    head_softmax<<<dim3(NR), dim3(256), 0, stream>>>(Zhead, LDHEAD, out, g0g1, lengths);

    // cluster 0: mid0 = hs @ t0_proj^T [2000,128]; logits -> out cols [2000,10000)
    gemm_wmma_f32<0, 1><<<dim3(1, cdiv(NR, 32), 1), gblk, 0, stream>>>(
        hs, H_, 0, t0_proj, H_, 0, nullptr,
        mid0, 128, 0, NR, 128, H_, 128);
    gemm_wmma_f32<0, 1><<<dim3(cdiv(C1_ - C0_, 128), cdiv(NR, 32), 1), gblk, 0, stream>>>(
        mid0, 128, 0, t0_out, 128, 0, nullptr,
        out + C0_, V_, 0, NR, C1_ - C0_, 128, C1_ - C0_);

    // cluster 1: mid1 = hs @ t1_proj^T [2000,32]; logits -> out cols [10000,32000)
    gemm_wmma_f32<0, 1><<<dim3(1, cdiv(NR, 32), 1), gblk, 0, stream>>>(
        hs, H_, 0, t1_proj, H_, 0, nullptr,
        mid1, 32, 0, NR, 32, H_, 32);
    gemm_wmma_f32<0, 1><<<dim3(cdiv(V_ - C1_, 128), cdiv(NR, 32), 1), gblk, 0, stream>>>(
        mid1, 32, 0, t1_out, 32, 0, nullptr,
        out + C1_, V_, 0, NR, V_ - C1_, 32, V_ - C1_);

    // final per-row log-softmax on the two tail segments (adds gate, masks rows)
    seg_softmax<<<dim3(NR), dim3(256), 0, stream>>>(out, C0_, C1_ - C0_, g0g1, 0, lengths);
    seg_softmax<<<dim3(NR), dim3(256), 0, stream>>>(out, C1_, V_ - C1_, g0g1, 1, lengths);
}